// MOLT_55645596287643
// MI455X (gfx1250) — compile-verified
//
#include <hip/hip_runtime.h>

#define D_MODEL 2048
#define BATCH   2048
#define NT      248
#define NGROUPS 5

typedef __attribute__((ext_vector_type(2))) float v2f;
typedef __attribute__((ext_vector_type(8))) float v8f;

__device__ __constant__ int c_cumn[NGROUPS + 1] = {0, 8, 24, 56, 120, 248};
__device__ __constant__ int c_rank[NGROUPS]     = {512, 256, 128, 64, 32};

struct GroupPtrs {
    const float* v[NGROUPS];
    const float* u[NGROUPS];
};

// ---------------------------------------------------------------- zero output
__global__ __launch_bounds__(256) void k_zero(float* __restrict__ out, int n) {
    int i = (blockIdx.x * 256 + threadIdx.x) * 4;
    if (i < n) {
        float4 z = make_float4(0.f, 0.f, 0.f, 0.f);
        *reinterpret_cast<float4*>(out + i) = z;
    }
}

// ------------------------------------------------------- gate = jumprelu(x e^T - b)
// grid.x = BATCH/16, block = 256.  Each block: 16 batch rows x all n columns.
__global__ __launch_bounds__(256) void k_gate(const float* __restrict__ x,
                                              const float* __restrict__ enc,
                                              const float* __restrict__ bias,
                                              float* __restrict__ gate,
                                              int n, int off) {
    __shared__ float lsx[16 * 65];
    __shared__ float lse[128 * 65];
    const int tid = threadIdx.x;
    const int bBase = blockIdx.x * 16;
    const int nOut = 16 * n;

    float acc[8];
#pragma unroll
    for (int a = 0; a < 8; ++a) acc[a] = 0.f;

    for (int k0 = 0; k0 < D_MODEL; k0 += 64) {
        for (int e = tid; e < 16 * 64; e += 256) {
            int row = e >> 6, kk = e & 63;
            lsx[row * 65 + kk] = x[(size_t)(bBase + row) * D_MODEL + k0 + kk];
        }
        for (int e = tid; e < n * 64; e += 256) {
            int row = e >> 6, kk = e & 63;
            lse[row * 65 + kk] = enc[(size_t)row * D_MODEL + k0 + kk];
        }
        __syncthreads();
#pragma unroll
        for (int rep = 0; rep < 8; ++rep) {
            int o = tid + rep * 256;
            if (o < nOut) {
                int row = o & 15, col = o >> 4;
                float s = acc[rep];
#pragma unroll 8
                for (int kk = 0; kk < 64; ++kk)
                    s = fmaf(lsx[row * 65 + kk], lse[col * 65 + kk], s);
                acc[rep] = s;
            }
        }
        __syncthreads();
    }
#pragma unroll
    for (int rep = 0; rep < 8; ++rep) {
        int o = tid + rep * 256;
        if (o < nOut) {
            int row = o & 15, col = o >> 4;
            float pre = acc[rep] - bias[col];
            float g = pre > 0.f ? pre : 0.f;   // jumprelu, threshold 0
            gate[(size_t)(bBase + row) * NT + off + col] = g;
        }
    }
}

// ---------------------------------------------- per-transform column sums / counts
__global__ __launch_bounds__(256) void k_colsum(const float* __restrict__ gate,
                                                float* __restrict__ colsum,
                                                float* __restrict__ poscnt) {
    __shared__ float ss[256];
    __shared__ float sc[256];
    const int t = blockIdx.x, tid = threadIdx.x;
    float s = 0.f, c = 0.f;
    for (int b = tid; b < BATCH; b += 256) {
        float g = gate[(size_t)b * NT + t];
        s += g;
        c += (g > 0.f) ? 1.f : 0.f;
    }
    ss[tid] = s; sc[tid] = c;
    __syncthreads();
    for (int w = 128; w > 0; w >>= 1) {
        if (tid < w) { ss[tid] += ss[tid + w]; sc[tid] += sc[tid + w]; }
        __syncthreads();
    }
    if (tid == 0) { colsum[t] = ss[0]; poscnt[t] = sc[0]; }
}

// ---------------------------------------------- Frobenius norm^2 of each U_i / V_i
// grid.x = 2*NT : blocks [0,248) -> V, [248,496) -> U
__global__ __launch_bounds__(256) void k_normsq(GroupPtrs p,
                                                float* __restrict__ vsq,
                                                float* __restrict__ usq) {
    const int b = blockIdx.x;
    const int isU = b >= NT;
    const int t = isU ? b - NT : b;
    int gi = 0;
    while (t >= c_cumn[gi + 1]) gi++;
    const int i = t - c_cumn[gi];
    const long len = (long)c_rank[gi] * D_MODEL;
    const float* base = (isU ? p.u[gi] : p.v[gi]) + (long)i * len;

    const int tid = threadIdx.x;
    float s = 0.f;
    for (long e = (long)tid * 4; e < len; e += 256 * 4) {
        float4 q = *reinterpret_cast<const float4*>(base + e);
        s += q.x * q.x + q.y * q.y + q.z * q.z + q.w * q.w;
    }
    __shared__ float ss[256];
    ss[tid] = s;
    __syncthreads();
    for (int w = 128; w > 0; w >>= 1) {
        if (tid < w) ss[tid] += ss[tid + w];
        __syncthreads();
    }
    if (tid == 0) (isU ? usq : vsq)[t] = ss[0];
}

// ------------------------------------------------ GEMM1: T = gated( X @ Vcat^T )
// X: (2048 x 2048) rm, Vcat: (4096 x 2048) rm  -> T: (2048 x 4096) rm
// grid = (BATCH/128, 4096/256), block = 256 (8 waves); wave tile = 64x64.
__global__ __launch_bounds__(256) void k_gemm1(const float* __restrict__ x,
                                               const float* __restrict__ vmat,
                                               const float* __restrict__ gate,
                                               float* __restrict__ T,
                                               int r, int off) {
    __shared__ float lsA[128 * 33];
    __shared__ float lsB[256 * 33];
    const int tid = threadIdx.x;
    const int lane = tid & 31, wv = tid >> 5;
    const int wm = wv & 1, wn = wv >> 1;          // wave tile: 64(M) x 64(N)
    const int l16 = lane & 15, lh = lane >> 4;
    const int mBase = blockIdx.x * 128;
    const int nBase = blockIdx.y * 256;

    v8f c[4][4];
#pragma unroll
    for (int tm = 0; tm < 4; ++tm)
#pragma unroll
        for (int tn = 0; tn < 4; ++tn)
            c[tm][tn] = (v8f){0.f, 0.f, 0.f, 0.f, 0.f, 0.f, 0.f, 0.f};

    for (int k0 = 0; k0 < D_MODEL; k0 += 32) {
        const bool pf = (k0 + 32) < D_MODEL;
#pragma unroll
        for (int rep = 0; rep < 4; ++rep) {        // A tile: 128 x 32
            int f = tid + rep * 256;               // [0,1024)
            int row = f >> 3, col = (f & 7) * 4;
            const float* ap = x + (size_t)(mBase + row) * D_MODEL + k0 + col;
            float4 qa = *reinterpret_cast<const float4*>(ap);
            lsA[row * 33 + col + 0] = qa.x; lsA[row * 33 + col + 1] = qa.y;
            lsA[row * 33 + col + 2] = qa.z; lsA[row * 33 + col + 3] = qa.w;
            if (pf && col == 0) __builtin_prefetch(ap + 32, 0, 0);
        }
#pragma unroll
        for (int rep = 0; rep < 8; ++rep) {        // B tile: 256 x 32
            int f = tid + rep * 256;               // [0,2048)
            int row = f >> 3, col = (f & 7) * 4;
            const float* bp = vmat + (size_t)(nBase + row) * D_MODEL + k0 + col;
            float4 qb = *reinterpret_cast<const float4*>(bp);
            lsB[row * 33 + col + 0] = qb.x; lsB[row * 33 + col + 1] = qb.y;
            lsB[row * 33 + col + 2] = qb.z; lsB[row * 33 + col + 3] = qb.w;
            if (pf && col == 0) __builtin_prefetch(bp + 32, 0, 0);
        }
        __syncthreads();
#pragma unroll
        for (int kk = 0; kk < 32; kk += 4) {
            const int kL = kk + 2 * lh;            // ISA 16x4 f32 A/B layout
            v2f a[4], bf[4];
#pragma unroll
            for (int tm = 0; tm < 4; ++tm) {
                int m = wm * 64 + tm * 16 + l16;
                a[tm].x = lsA[m * 33 + kL];
                a[tm].y = lsA[m * 33 + kL + 1];
            }
#pragma unroll
            for (int tn = 0; tn < 4; ++tn) {
                int nn = wn * 64 + tn * 16 + l16;
                bf[tn].x = lsB[nn * 33 + kL];
                bf[tn].y = lsB[nn * 33 + kL + 1];
            }
#pragma unroll
            for (int tm = 0; tm < 4; ++tm)
#pragma unroll
                for (int tn = 0; tn < 4; ++tn)
                    c[tm][tn] = __builtin_amdgcn_wmma_f32_16x16x4_f32(
                        false, a[tm], false, bf[tn], (short)0, c[tm][tn],
                        false, false);
        }
        __syncthreads();
    }
    // epilogue: multiply by gate[b, transform(col)] and store to T
#pragma unroll
    for (int tm = 0; tm < 4; ++tm)
#pragma unroll
        for (int tn = 0; tn < 4; ++tn) {
            int colT = nBase + wn * 64 + tn * 16;  // tile lies in one transform (r>=32)
            int ti = colT / r;
            int col = colT + l16;
#pragma unroll
            for (int v = 0; v < 8; ++v) {
                int row = mBase + wm * 64 + tm * 16 + v + 8 * lh;
                float g = gate[(size_t)row * NT + off + ti];
                T[(size_t)row * 4096 + col] = c[tm][tn][v] * g;
            }
        }
}

// ------------------------------------------------ GEMM2: OUT += T @ Ucat
// T: (2048 x 4096) rm;  Ucat(k= i*r+j, dd) = U[i*d*r + dd*r + j]
// grid = (BATCH/128, D_MODEL/256), block = 256; wave tile = 64x64.
__global__ __launch_bounds__(256) void k_gemm2(const float* __restrict__ T,
                                               const float* __restrict__ u,
                                               float* __restrict__ out,
                                               int r) {
    __shared__ float lsA[128 * 33];
    __shared__ float lsB[256 * 33];
    const int tid = threadIdx.x;
    const int lane = tid & 31, wv = tid >> 5;
    const int wm = wv & 1, wn = wv >> 1;
    const int l16 = lane & 15, lh = lane >> 4;
    const int mBase = blockIdx.x * 128;
    const int nBase = blockIdx.y * 256;

    v8f c[4][4];
#pragma unroll
    for (int tm = 0; tm < 4; ++tm)
#pragma unroll
        for (int tn = 0; tn < 4; ++tn)
            c[tm][tn] = (v8f){0.f, 0.f, 0.f, 0.f, 0.f, 0.f, 0.f, 0.f};

    for (int k0 = 0; k0 < 4096; k0 += 32) {
        const int i = k0 / r;                      // 32-chunks never straddle a transform
        const int j0 = k0 - i * r;
        const float* ub = u + (size_t)i * D_MODEL * r + j0;
        const bool pf = (k0 + 32) < 4096;
        const int in_ = (k0 + 32) / r;
        const float* ubn = u + (size_t)in_ * D_MODEL * r + ((k0 + 32) - in_ * r);
#pragma unroll
        for (int rep = 0; rep < 4; ++rep) {        // A tile from T: 128 x 32
            int f = tid + rep * 256;
            int row = f >> 3, col = (f & 7) * 4;
            const float* ap = T + (size_t)(mBase + row) * 4096 + k0 + col;
            float4 qa = *reinterpret_cast<const float4*>(ap);
            lsA[row * 33 + col + 0] = qa.x; lsA[row * 33 + col + 1] = qa.y;
            lsA[row * 33 + col + 2] = qa.z; lsA[row * 33 + col + 3] = qa.w;
            if (pf && col == 0) __builtin_prefetch(ap + 32, 0, 0);
        }
#pragma unroll
        for (int rep = 0; rep < 8; ++rep) {        // B tile from U: 256 x 32 (stride r)
            int f = tid + rep * 256;
            int row = f >> 3, col = (f & 7) * 4;
            const float* bp = ub + (size_t)(nBase + row) * r + col;
            float4 qb = *reinterpret_cast<const float4*>(bp);
            lsB[row * 33 + col + 0] = qb.x; lsB[row * 33 + col + 1] = qb.y;
            lsB[row * 33 + col + 2] = qb.z; lsB[row * 33 + col + 3] = qb.w;
            if (pf && col == 0)
                __builtin_prefetch(ubn + (size_t)(nBase + row) * r, 0, 0);
        }
        __syncthreads();
#pragma unroll
        for (int kk = 0; kk < 32; kk += 4) {
            const int kL = kk + 2 * lh;
            v2f a[4], bf[4];
#pragma unroll
            for (int tm = 0; tm < 4; ++tm) {
                int m = wm * 64 + tm * 16 + l16;
                a[tm].x = lsA[m * 33 + kL];
                a[tm].y = lsA[m * 33 + kL + 1];
            }
#pragma unroll
            for (int tn = 0; tn < 4; ++tn) {
                int nn = wn * 64 + tn * 16 + l16;
                bf[tn].x = lsB[nn * 33 + kL];
                bf[tn].y = lsB[nn * 33 + kL + 1];
            }
#pragma unroll
            for (int tm = 0; tm < 4; ++tm)
#pragma unroll
                for (int tn = 0; tn < 4; ++tn)
                    c[tm][tn] = __builtin_amdgcn_wmma_f32_16x16x4_f32(
                        false, a[tm], false, bf[tn], (short)0, c[tm][tn],
                        false, false);
        }
        __syncthreads();
    }
#pragma unroll
    for (int tm = 0; tm < 4; ++tm)
#pragma unroll
        for (int tn = 0; tn < 4; ++tn) {
            int col = nBase + wn * 64 + tn * 16 + l16;
#pragma unroll
            for (int v = 0; v < 8; ++v) {
                int row = mBase + wm * 64 + tm * 16 + v + 8 * lh;
                float* p = out + (size_t)row * D_MODEL + col;
                *p += c[tm][tn][v];                // groups run sequentially on stream
            }
        }
}

// ------------------------------------------------------------- scalar statistics
__global__ __launch_bounds__(256) void k_finalize(const float* __restrict__ colsum,
                                                  const float* __restrict__ poscnt,
                                                  const float* __restrict__ vsq,
                                                  const float* __restrict__ usq,
                                                  float* __restrict__ stats) {
    __shared__ float s1[256];
    __shared__ float s2[256];
    const int tid = threadIdx.x;
    float sp = 0.f, ac = 0.f;
    if (tid < NT) {
        int gi = 0;
        while (tid >= c_cumn[gi + 1]) gi++;
        float rk = (float)c_rank[gi];
        float mean_abs = colsum[tid] / (float)BATCH;      // gate >= 0
        float frob = sqrtf(usq[tid] * vsq[tid]) * rsqrtf((float)D_MODEL * rk);
        sp = tanhf(mean_abs) * frob;
        ac = poscnt[tid];
    }
    s1[tid] = sp; s2[tid] = ac;
    __syncthreads();
    for (int w = 128; w > 0; w >>= 1) {
        if (tid < w) { s1[tid] += s1[tid + w]; s2[tid] += s2[tid + w]; }
        __syncthreads();
    }
    if (tid == 0) {
        stats[0] = s1[0];                  // total_sparsity
        stats[1] = s2[0] / (float)BATCH;   // total_active
    }
}

// ------------------------------------------------------------------------ host
extern "C" void kernel_launch(void* const* d_in, const int* in_sizes, int n_in,
                              void* d_out, int out_size, void* d_ws, size_t ws_size,
                              hipStream_t stream) {
    (void)in_sizes; (void)n_in; (void)out_size; (void)ws_size;
    const float* x = (const float*)d_in[0];
    const float* V[NGROUPS]; const float* U[NGROUPS];
    const float* E[NGROUPS]; const float* Bv[NGROUPS];
    for (int g = 0; g < NGROUPS; ++g) {
        V[g]  = (const float*)d_in[1 + g];
        U[g]  = (const float*)d_in[6 + g];
        E[g]  = (const float*)d_in[11 + g];
        Bv[g] = (const float*)d_in[16 + g];
    }
    float* out = (float*)d_out;

    float* ws     = (float*)d_ws;
    float* T      = ws;                                  // 2048*4096
    float* gate   = ws + (size_t)BATCH * 4096;           // 2048*248
    float* colsum = gate + (size_t)BATCH * NT;           // 248
    float* poscnt = colsum + NT;                         // 248
    float* vsq    = poscnt + NT;                         // 248
    float* usq    = vsq + NT;                            // 248

    static const int ns[NGROUPS]   = {8, 16, 32, 64, 128};
    static const int rs[NGROUPS]   = {512, 256, 128, 64, 32};
    static const int offs[NGROUPS] = {0, 8, 24, 56, 120};

    k_zero<<<(BATCH * D_MODEL) / (256 * 4), 256, 0, stream>>>(out, BATCH * D_MODEL);

    for (int g = 0; g < NGROUPS; ++g)
        k_gate<<<BATCH / 16, 256, 0, stream>>>(x, E[g], Bv[g], gate, ns[g], offs[g]);

    k_colsum<<<NT, 256, 0, stream>>>(gate, colsum, poscnt);

    GroupPtrs p;
    for (int g = 0; g < NGROUPS; ++g) { p.v[g] = V[g]; p.u[g] = U[g]; }
    k_normsq<<<2 * NT, 256, 0, stream>>>(p, vsq, usq);

    for (int g = 0; g < NGROUPS; ++g) {
        k_gemm1<<<dim3(BATCH / 128, 4096 / 256), 256, 0, stream>>>(
            x, V[g], gate, T, rs[g], offs[g]);
        k_gemm2<<<dim3(BATCH / 128, D_MODEL / 256), 256, 0, stream>>>(
            T, U[g], out, rs[g]);
    }

    k_finalize<<<1, 256, 0, stream>>>(colsum, poscnt, vsq, usq,
                                      out + (size_t)BATCH * D_MODEL);
}